// LR_GCN_33397665694594
// MI455X (gfx1250) — compile-verified
//
#include <hip/hip_runtime.h>
#include <hip/hip_bf16.h>

typedef float v2f __attribute__((ext_vector_type(2)));
typedef float v8f __attribute__((ext_vector_type(8)));

// ---------------------------------------------------------------------------
// Zero-init workspace (h1 + h2 aggregation buffers).
// ---------------------------------------------------------------------------
__global__ void lrgcn_zero_kernel(unsigned* __restrict__ p, long long n) {
    long long i = (long long)blockIdx.x * blockDim.x + threadIdx.x;
    long long stride = (long long)gridDim.x * blockDim.x;
    for (; i < n; i += stride) p[i] = 0u;
}

// ---------------------------------------------------------------------------
// One LRConv layer. F = per-node feature width (4 for conv1, 16 for conv2).
// One wave32 processes a tile of 16 edges:
//   m = relu(relu([feat[src] || feat[dst]] @ W1 + b1) @ W2 + b2)   (16x16 tile)
//   outb[src*16 + n] = max(outb[...], m)  via u32 atomicMax (m >= 0).
// All matmuls use exact-fp32 V_WMMA_F32_16X16X4_F32.
// ---------------------------------------------------------------------------
template <int F>
__global__ __launch_bounds__(256) void lrconv_wmma_kernel(
    const float* __restrict__ feat, const long long* __restrict__ ei,
    const float* __restrict__ w1, const float* __restrict__ b1,
    const float* __restrict__ w2, const float* __restrict__ b2,
    unsigned* __restrict__ outb, int E, int ntiles) {

    constexpr int HALF = F / 4;           // K-chunks per endpoint
    __shared__ float xpose[8][256];       // per-wave 16x16 transpose staging

    const int wave = threadIdx.x >> 5;
    const int lane = threadIdx.x & 31;
    const int hi   = lane >> 4;           // half-wave select (k+0/1 vs k+2/3)
    const int r    = lane & 15;           // row (M) for A, col (N) for B/C/D

    int t = blockIdx.x * (blockDim.x >> 5) + wave;
    if (t >= ntiles) t = ntiles - 1;      // clamp (idempotent under max-agg)
    const int te = t * 16;

    int erow = te + r;                    // this lane's edge row of the A tile
    if (erow >= E) erow = E - 1;
    const long long srcA = ei[erow];          // edge_index[0][e]
    const long long dstA = ei[E + erow];      // edge_index[1][e]

    // ---- layer 1: [16 x F*2] @ [F*2 x 16] via 2*HALF chained K=4 WMMAs ----
    v8f acc = {};
    #pragma unroll
    for (int c = 0; c < 2 * HALF; ++c) {
        const long long node = (c < HALF) ? srcA : dstA;
        const int kb = (c % HALF) * 4 + 2 * hi;
        const float* fp = feat + node * F + kb;
        v2f a; a.x = fp[0]; a.y = fp[1];
        const int kw = 4 * c + 2 * hi;
        v2f b; b.x = w1[kw * 16 + r]; b.y = w1[(kw + 1) * 16 + r];
        acc = __builtin_amdgcn_wmma_f32_16x16x4_f32(
                  false, a, false, b, (short)0, acc, false, false);
    }

    // bias + relu, then stage C/D-layout tile into LDS row-major [edge][hid]
    const float b1r = b1[r];
    #pragma unroll
    for (int v = 0; v < 8; ++v) {
        float h = fmaxf(acc[v] + b1r, 0.0f);
        xpose[wave][(v + 8 * hi) * 16 + r] = h;
    }
    __syncthreads();   // uniform across block (tile index is clamped above)

    // ---- layer 2: [16 x 16] @ [16 x 16] via 4 chained K=4 WMMAs ----------
    v8f acc2 = {};
    #pragma unroll
    for (int c = 0; c < 4; ++c) {
        const int kb = 4 * c + 2 * hi;
        v2f a; a.x = xpose[wave][r * 16 + kb];
               a.y = xpose[wave][r * 16 + kb + 1];
        v2f b; b.x = w2[kb * 16 + r]; b.y = w2[(kb + 1) * 16 + r];
        acc2 = __builtin_amdgcn_wmma_f32_16x16x4_f32(
                   false, a, false, b, (short)0, acc2, false, false);
    }

    // bias + relu + scatter-max at src (u32 max == f32 max for values >= 0)
    const float b2r = b2[r];
    #pragma unroll
    for (int v = 0; v < 8; ++v) {
        int el = te + v + 8 * hi;
        if (el >= E) el = E - 1;
        const long long s = ei[el];
        const float m = fmaxf(acc2[v] + b2r, 0.0f);
        atomicMax(&outb[s * 16 + r], __float_as_uint(m));
    }
}

// ---------------------------------------------------------------------------
// Node head: out = sigmoid(relu(h @ l1_w + l1_b) @ l2_w + l2_b)
// ---------------------------------------------------------------------------
__global__ void lrgcn_head_kernel(const float* __restrict__ h2,
                                  const float* __restrict__ l1w,
                                  const float* __restrict__ l1b,
                                  const float* __restrict__ l2w,
                                  const float* __restrict__ l2b,
                                  float* __restrict__ out, int n) {
    int i = blockIdx.x * blockDim.x + threadIdx.x;
    if (i >= n) return;
    float h[16];
    #pragma unroll
    for (int k = 0; k < 16; ++k) h[k] = h2[(long long)i * 16 + k];
    float z = l2b[0];
    #pragma unroll
    for (int c = 0; c < 16; ++c) {
        float t = l1b[c];
        #pragma unroll
        for (int k = 0; k < 16; ++k) t += h[k] * l1w[k * 16 + c];
        z += fmaxf(t, 0.0f) * l2w[c];
    }
    out[i] = 1.0f / (1.0f + __expf(-z));
}

// ---------------------------------------------------------------------------
extern "C" void kernel_launch(void* const* d_in, const int* in_sizes, int n_in,
                              void* d_out, int out_size, void* d_ws, size_t ws_size,
                              hipStream_t stream) {
    const float*     x    = (const float*)d_in[0];
    const long long* ei   = (const long long*)d_in[1];   // int64 [2, E]
    const float* c1w1 = (const float*)d_in[2];
    const float* c1b1 = (const float*)d_in[3];
    const float* c1w2 = (const float*)d_in[4];
    const float* c1b2 = (const float*)d_in[5];
    const float* c2w1 = (const float*)d_in[6];
    const float* c2b1 = (const float*)d_in[7];
    const float* c2w2 = (const float*)d_in[8];
    const float* c2b2 = (const float*)d_in[9];
    const float* l1w  = (const float*)d_in[10];
    const float* l1b  = (const float*)d_in[11];
    const float* l2w  = (const float*)d_in[12];
    const float* l2b  = (const float*)d_in[13];

    const int N = in_sizes[0] / 4;   // nodes (x is [N,4])
    const int E = in_sizes[1] / 2;   // edges (edge_index is [2,E])

    unsigned* h1 = (unsigned*)d_ws;                         // [N,16] fp32-as-u32
    unsigned* h2 = h1 + (size_t)N * 16;                     // [N,16]

    // zero both aggregation buffers (max identity for non-negative messages)
    const long long zn = (long long)N * 32;
    int zblk = (int)((zn + 255) / 256);
    if (zblk > 65535) zblk = 65535;
    lrgcn_zero_kernel<<<zblk, 256, 0, stream>>>((unsigned*)d_ws, zn);

    const int ntiles = (E + 15) / 16;
    const int cblk   = (ntiles + 7) / 8;   // 8 waves (tiles) per 256-thread block

    lrconv_wmma_kernel<4><<<cblk, 256, 0, stream>>>(
        x, ei, c1w1, c1b1, c1w2, c1b2, h1, E, ntiles);

    lrconv_wmma_kernel<16><<<cblk, 256, 0, stream>>>(
        (const float*)h1, ei, c2w1, c2b1, c2w2, c2b2, h2, E, ntiles);

    lrgcn_head_kernel<<<(N + 255) / 256, 256, 0, stream>>>(
        (const float*)h2, l1w, l1b, l2w, l2b, (float*)d_out, N);
}